// P_Pooling_Module_20340965114066
// MI455X (gfx1250) — compile-verified
//
#include <hip/hip_runtime.h>
#include <cstdint>
#include <cstddef>

// Problem geometry (fixed by the reference's setup_inputs)
#define B_  8
#define C_  256
#define H_  128
#define W_  128
#define HO_ 64
#define WO_ 64
#define PLANE (H_ * W_)      // 16384 floats = 64 KB
#define OPLANE (HO_ * WO_)   // 4096

__global__ __launch_bounds__(256) void ppool_fused_kernel(
    const float* __restrict__ bottom,
    const float* __restrict__ w1,   // (C,1,3,3)
    const float* __restrict__ a1,   // (C,)
    const float* __restrict__ w2,   // (C,)
    const float* __restrict__ b2,   // (C,)
    float* __restrict__ res,        // (B,C,64,64)
    float* __restrict__ pexp)       // (B,C,64,64)
{
    __shared__ float smem[PLANE];   // one full 128x128 channel plane
    __shared__ float red[8];        // per-wave partial maxima

    const int bc   = blockIdx.x;        // b*C + c
    const int c    = bc & (C_ - 1);
    const int tid  = threadIdx.x;       // 0..255
    const int lane = tid & 31;
    const int wave = tid >> 5;          // 0..7

    // ---------------------------------------------------------------
    // Stage 1: async-copy the plane global -> LDS (CDNA5 async path).
    // Wave w copies the contiguous 8 KB stripe [w*8K, w*8K+8K):
    // 16 rounds x 32 lanes x 16 B. Layout identical in LDS & global,
    // so after s_wait_asynccnt 0 this wave's own stripe is LDS-visible
    // to itself without a workgroup barrier.
    // ---------------------------------------------------------------
    {
        const float* gsrc = bottom + (size_t)bc * PLANE;
        const unsigned long long gaddr = (unsigned long long)(uintptr_t)gsrc;
        const unsigned lds_base   = (unsigned)(uintptr_t)(&smem[0]); // low 32b = LDS offset
        const unsigned stripe     = (unsigned)wave * 8192u;          // bytes
        #pragma unroll
        for (int i = 0; i < 16; ++i) {
            unsigned off  = stripe + ((unsigned)(lane + i * 32) * 16u); // 16B aligned
            unsigned ldsa = lds_base + off;
            asm volatile("global_load_async_to_lds_b128 %0, %1, %2"
                         :
                         : "v"(ldsa), "v"(off), "s"(gaddr)
                         : "memory");
        }
        asm volatile("s_wait_asynccnt 0x0" ::: "memory");
    }

    // ---------------------------------------------------------------
    // Stage 2a: each wave max-reduces |x| over its own stripe (the
    // exact bytes it just copied) -> no barrier needed yet.
    // Lane-contiguous dword stripes -> conflict-free LDS banks.
    // ---------------------------------------------------------------
    {
        const int base = wave * 2048;   // dwords: stripe/4
        float vmax = 0.0f;
        #pragma unroll
        for (int i = 0; i < 2048 / 32; ++i)
            vmax = fmaxf(vmax, fabsf(smem[base + lane + i * 32]));
        #pragma unroll
        for (int off = 16; off > 0; off >>= 1)
            vmax = fmaxf(vmax, __shfl_xor(vmax, off, 32));
        if (lane == 0) red[wave] = vmax;
    }
    __syncthreads();   // all stripes copied + all partial maxima published

    float m = red[0];
    #pragma unroll
    for (int w = 1; w < 8; ++w) m = fmaxf(m, red[w]);
    m += 1.0f;
    const float inv_m = 1.0f / m;

    // Per-channel parameters (uniform across the workgroup -> scalar loads).
    float wk[9];
    #pragma unroll
    for (int k = 0; k < 9; ++k) wk[k] = w1[c * 9 + k];
    const float a1c = a1[c];
    const float w2c = w2[c];
    const float b2c = b2[c];

    // ---------------------------------------------------------------
    // Stage 3+4 fused: the 3x3/stride-2/pad-1 conv window and the
    // pooling window are identical per output pixel -> read the 9
    // taps once into registers, compute exponent p, then the p-mean.
    // Only top/left padding can occur (max tap index = 2*63+1 = 127).
    // ---------------------------------------------------------------
    const size_t obase = (size_t)bc * OPLANE;
    for (int i = 0; i < OPLANE / 256; ++i) {
        const int pos = tid + i * 256;
        const int oy  = pos >> 6;
        const int ox  = pos & 63;
        const int iy0 = 2 * oy - 1;
        const int ix0 = 2 * ox - 1;

        float t[9];
        float acc = 0.0f;
        #pragma unroll
        for (int ky = 0; ky < 3; ++ky) {
            const int iy = iy0 + ky;
            #pragma unroll
            for (int kx = 0; kx < 3; ++kx) {
                const int ix = ix0 + kx;
                const int k  = ky * 3 + kx;
                const bool ok = (iy >= 0) && (ix >= 0);
                const float v = ok ? smem[iy * W_ + ix] : 0.0f;
                t[k] = v;
                acc  = fmaf(wk[k], v, acc);   // conv on un-normalized taps
            }
        }
        // normalize factored out of the conv sum: conv(x/m) = inv_m * conv(x)
        const float vconv = acc * inv_m;
        const float pre   = (vconv > 0.0f) ? vconv : a1c * vconv;   // PReLU
        float p = fmaf(pre, w2c, b2c);                              // 1x1 dw conv + bias
        p = fminf(fmaxf(p, 1.0f), 110.0f);                          // Hardtanh [min_p, max_p]

        // p-mean pool: (mean relu(t)^p + eps)^(1/p)
        float s = 0.0f;
        #pragma unroll
        for (int k = 0; k < 9; ++k) {
            const float r = fmaxf(t[k], 0.0f);
            // r == 0: log->-inf, exp->0 (correct: 0^p = 0 for p>=1)
            s += __expf(__logf(r) * p);
        }
        const float mean = s * (1.0f / 9.0f);
        const float out  = __expf(__logf(mean + 1e-12f) / p);

        // Outputs are written once, never re-read -> non-temporal stores
        // (TH_STORE_NT) so they don't displace the streaming input in L2.
        __builtin_nontemporal_store(out, &res[obase + pos]);
        __builtin_nontemporal_store(p,   &pexp[obase + pos]);
    }
}

extern "C" void kernel_launch(void* const* d_in, const int* in_sizes, int n_in,
                              void* d_out, int out_size, void* d_ws, size_t ws_size,
                              hipStream_t stream) {
    (void)in_sizes; (void)n_in; (void)out_size; (void)d_ws; (void)ws_size;
    const float* bottom = (const float*)d_in[0];
    const float* w1     = (const float*)d_in[1];
    const float* a1     = (const float*)d_in[2];
    const float* w2     = (const float*)d_in[3];
    const float* b2     = (const float*)d_in[4];

    float* res  = (float*)d_out;
    float* pexp = (float*)d_out + (size_t)B_ * C_ * OPLANE;  // tuple order: (res, pexp)

    dim3 grid(B_ * C_);   // one workgroup per (b, c) plane
    dim3 block(256);      // 8 wave32s
    hipLaunchKernelGGL(ppool_fused_kernel, grid, block, 0, stream,
                       bottom, w1, a1, w2, b2, res, pexp);
}